// EGCL_29927332118823
// MI455X (gfx1250) — compile-verified
//
#include <hip/hip_runtime.h>

#define NNODES 50000
#define NEDGES 800000

typedef __attribute__((ext_vector_type(2))) float v2f;
typedef __attribute__((ext_vector_type(8))) float v8f;

__device__ __forceinline__ v8f wmma_f32_k4(v2f a, v2f b, v8f c) {
  // D(16x16,f32) += A(16x4,f32) x B(4x16,f32)
  return __builtin_amdgcn_wmma_f32_16x16x4_f32(false, a, false, b, (short)0, c,
                                               false, false);
}

__device__ __forceinline__ v8f zero8() {
  v8f z = {0.f, 0.f, 0.f, 0.f, 0.f, 0.f, 0.f, 0.f};
  return z;
}

__device__ __forceinline__ float sigm(float x) { return 1.0f / (1.0f + __expf(-x)); }

// ---------------- compile-time Clifford algebra tables ----------------
constexpr int popc3(int x) { return (x & 1) + ((x >> 1) & 1) + ((x >> 2) & 1); }
constexpr float csign(int a, int b) {
  int s = 0, aa = a >> 1;
  while (aa) { s += popc3(aa & b); aa >>= 1; }
  return (s & 1) ? -1.0f : 1.0f;
}
constexpr bool pex(int gi, int gj, int gk) {
  for (int a = 0; a < 8; a++)
    for (int b = 0; b < 8; b++)
      if (popc3(a) == gi && popc3(b) == gk && popc3(a ^ b) == gj) return true;
  return false;
}
constexpr int pathIdx(int gi, int gj, int gk) {
  int n = 0;
  for (int i = 0; i < 4; i++)
    for (int j = 0; j < 4; j++)
      for (int k = 0; k < 4; k++) {
        bool e = pex(i, j, k);
        if (i == gi && j == gj && k == gk) return e ? n : -1;
        if (e) n++;
      }
  return -1;
}
struct GPTab { int j[64]; float s[64]; int p[64]; };
constexpr GPTab makeGP() {
  GPTab t{};
  const int M[8] = {0, 1, 2, 4, 3, 5, 6, 7};  // pos->mask
  const int P[8] = {0, 1, 2, 4, 3, 5, 6, 7};  // mask->pos
  for (int I = 0; I < 8; I++)
    for (int K = 0; K < 8; K++) {
      int a = M[I], b = M[K];
      t.j[I * 8 + K] = P[a ^ b];
      t.s[I * 8 + K] = csign(a, b);
      t.p[I * 8 + K] = pathIdx(popc3(a), popc3(a ^ b), popc3(b));
    }
  return t;
}
constexpr GPTab GPT = makeGP();
constexpr int GRADE[8] = {0, 1, 1, 1, 2, 2, 2, 3};

// workspace layout (floats)
#define AGG_FLOATS (NNODES * 128)
#define OFF_FRAGE  (AGG_FLOATS)
#define OFF_FRAGN  (OFF_FRAGE + 1024)
#define OFF_FRAGL  (OFF_FRAGN + 2048)
#define OFF_FRAGR  (OFF_FRAGL + 2048)
#define OFF_BLP    (OFF_FRAGR + 2048)
#define OFF_BRP    (OFF_BLP + 16)

// ---------------- pack kernel: A-fragment layouts + composed linears ----------------
// fragX[((g*CH+k)*32 + lane)*2 + v] = W[M=lane&15, K=4k+2*(lane>>4)+v, g]
__global__ __launch_bounds__(32)
void egcl_pack_kernel(const float* __restrict__ W_edge, const float* __restrict__ W_node,
                      const float* __restrict__ W_left, const float* __restrict__ W_right,
                      const float* __restrict__ b_node, const float* __restrict__ b_left,
                      float* __restrict__ ws) {
  const int lane = threadIdx.x;
  const int col = lane & 15, hi = lane >> 4;
  float* fragE = ws + OFF_FRAGE;
  float* fragN = ws + OFF_FRAGN;
  float* fragL = ws + OFF_FRAGL;
  float* fragR = ws + OFF_FRAGR;

  // W_edge fragments (16x16 per grade, 4 K-chunks)
  for (int g = 0; g < 4; ++g)
    for (int k = 0; k < 4; ++k)
      for (int v = 0; v < 2; ++v)
        fragE[((g * 4 + k) * 32 + lane) * 2 + v] =
            W_edge[col * 64 + (4 * k + 2 * hi + v) * 4 + g];

  // W_node fragments (16x32 per grade, 8 K-chunks) and composed
  // W_left*W_node, W_right*W_node fragments
  for (int g = 0; g < 4; ++g)
    for (int k = 0; k < 8; ++k)
      for (int v = 0; v < 2; ++v) {
        int m = 4 * k + 2 * hi + v;
        fragN[((g * 8 + k) * 32 + lane) * 2 + v] = W_node[col * 128 + m * 4 + g];
        float sl = 0.f, sr = 0.f;
        for (int n = 0; n < 16; ++n) {
          float wn = W_node[n * 128 + m * 4 + g];
          sl += W_left[col * 64 + n * 4 + g] * wn;
          sr += W_right[col * 64 + n * 4 + g] * wn;
        }
        fragL[((g * 8 + k) * 32 + lane) * 2 + v] = sl;
        fragR[((g * 8 + k) * 32 + lane) * 2 + v] = sr;
      }

  // bias images through the left/right linears (b_node lives in blade 0 = grade 0)
  if (lane < 16) {
    float sl = b_left[lane], sr = 0.f;
    for (int n = 0; n < 16; ++n) {
      sl += W_left[lane * 64 + n * 4 + 0] * b_node[n];
      sr += W_right[lane * 64 + n * 4 + 0] * b_node[n];
    }
    ws[OFF_BLP + lane] = sl;
    ws[OFF_BRP + lane] = sr;
  }
}

// ---------------- edge kernel: msg = silu(linear(h[r]-h[c])), atomic scatter ----
// 1 wave = 16 edges; each lane loads ITS OWN edge's channels -> no LDS at all.
__global__ __launch_bounds__(32)
void egcl_edge_kernel(const float* __restrict__ h, const int* __restrict__ ei,
                      const float* __restrict__ fragE, const float* __restrict__ b_edge,
                      const float* __restrict__ a_e, const float* __restrict__ b_e,
                      float* __restrict__ agg) {
  const int lane = threadIdx.x;
  const int col = lane & 15, hi = lane >> 4;
  const int e0 = blockIdx.x * 16;
  const int r = ei[e0 + col];
  const int c = ei[NEDGES + e0 + col];

  // A fragments: uniform, contiguous broadcast b64 loads
  v2f aw[4][4];
  #pragma unroll
  for (int g = 0; g < 4; ++g)
    #pragma unroll
    for (int k = 0; k < 4; ++k)
      aw[g][k] = *(const v2f*)(fragE + ((g * 4 + k) * 32 + lane) * 2);

  // diff of this lane's edge, channels ch = 4k + 2hi + v (vectorized b128 loads)
  float d[4][2][8];
  #pragma unroll
  for (int k = 0; k < 4; ++k)
    #pragma unroll
    for (int v = 0; v < 2; ++v) {
      int ch = 4 * k + 2 * hi + v;
      const float4* pr = (const float4*)(h + (size_t)r * 128 + ch * 8);
      const float4* pc = (const float4*)(h + (size_t)c * 128 + ch * 8);
      float4 a0 = pr[0], a1 = pr[1], c0 = pc[0], c1 = pc[1];
      d[k][v][0] = a0.x - c0.x; d[k][v][1] = a0.y - c0.y;
      d[k][v][2] = a0.z - c0.z; d[k][v][3] = a0.w - c0.w;
      d[k][v][4] = a1.x - c1.x; d[k][v][5] = a1.y - c1.y;
      d[k][v][6] = a1.z - c1.z; d[k][v][7] = a1.w - c1.w;
    }

  v8f acc[8];
  #pragma unroll
  for (int i = 0; i < 8; ++i) {
    v8f cacc = zero8();
    #pragma unroll
    for (int k = 0; k < 4; ++k) {
      v2f b;  // B[v][lane] = X[K=4k+v+2hi (ch), N=col (this lane's edge)]
      b.x = d[k][0][i];
      b.y = d[k][1][i];
      cacc = wmma_f32_k4(aw[GRADE[i]][k], b, cacc);
    }
    acc[i] = cacc;
  }

  #pragma unroll
  for (int v = 0; v < 8; ++v) {
    int ch = v + 8 * hi;  // D layout: M = v + 8*hi, N = col
    float x0 = acc[0][v] + b_edge[ch];
    float x1 = acc[1][v], x2 = acc[2][v], x3 = acc[3][v];
    float x4 = acc[4][v], x5 = acc[5][v], x6 = acc[6][v], x7 = acc[7][v];
    float i1 = x1 * x1 + x2 * x2 + x3 * x3;
    float i2 = x4 * x4 + x5 * x5 + x6 * x6;
    float i3 = x7 * x7;
    float g0 = sigm(a_e[ch * 4 + 0] * x0 + b_e[ch * 4 + 0]);
    float g1 = sigm(a_e[ch * 4 + 1] * i1 + b_e[ch * 4 + 1]);
    float g2 = sigm(a_e[ch * 4 + 2] * i2 + b_e[ch * 4 + 2]);
    float g3 = sigm(a_e[ch * 4 + 3] * i3 + b_e[ch * 4 + 3]);
    float* dst = agg + (size_t)r * 128 + ch * 8;  // r = this lane's edge row
    atomicAdd(dst + 0, g0 * x0);
    atomicAdd(dst + 1, g1 * x1);
    atomicAdd(dst + 2, g1 * x2);
    atomicAdd(dst + 3, g1 * x3);
    atomicAdd(dst + 4, g2 * x4);
    atomicAdd(dst + 5, g2 * x5);
    atomicAdd(dst + 6, g2 * x6);
    atomicAdd(dst + 7, g3 * x7);
  }
}

// ---------------- node kernel: nu/xl/xr in one pass -> gp -> silu -> residual ----
// 1 wave = 16 nodes; xcat lives in registers, no LDS.
__global__ __launch_bounds__(32)
void egcl_node_kernel(const float* __restrict__ h, const float* __restrict__ agg,
                      const float* __restrict__ fragN, const float* __restrict__ fragL,
                      const float* __restrict__ fragR, const float* __restrict__ b_node,
                      const float* __restrict__ blp, const float* __restrict__ brp,
                      const float* __restrict__ gp_w, const float* __restrict__ norm_a,
                      const float* __restrict__ a_n, const float* __restrict__ b_n,
                      float* __restrict__ out) {
  const int lane = threadIdx.x;
  const int col = lane & 15, hi = lane >> 4;
  const int n0 = blockIdx.x * 16;
  const int node = n0 + col;

  // xcat = concat(h, agg) for this lane's node, channels ch = 4k + 2hi + v
  float xc[8][2][8];
  #pragma unroll
  for (int k = 0; k < 8; ++k) {
    const float* base = (k < 4) ? h : agg;  // ch<16 -> h, ch>=16 -> agg
    #pragma unroll
    for (int v = 0; v < 2; ++v) {
      int chm = 4 * (k & 3) + 2 * hi + v;
      const float4* p = (const float4*)(base + (size_t)node * 128 + chm * 8);
      float4 q0 = p[0], q1 = p[1];
      xc[k][v][0] = q0.x; xc[k][v][1] = q0.y; xc[k][v][2] = q0.z; xc[k][v][3] = q0.w;
      xc[k][v][4] = q1.x; xc[k][v][5] = q1.y; xc[k][v][6] = q1.z; xc[k][v][7] = q1.w;
    }
  }

  // nu = Wn*xcat, xl = (Wl*Wn)*xcat, xr = (Wr*Wn)*xcat -- shared B fragments
  v8f nu[8], xl[8], xr[8];
  #pragma unroll
  for (int i = 0; i < 8; ++i) { nu[i] = zero8(); xl[i] = zero8(); xr[i] = zero8(); }

  #pragma unroll
  for (int k = 0; k < 8; ++k) {
    v2f an[4], al[4], ar[4];
    #pragma unroll
    for (int g = 0; g < 4; ++g) {
      an[g] = *(const v2f*)(fragN + ((g * 8 + k) * 32 + lane) * 2);
      al[g] = *(const v2f*)(fragL + ((g * 8 + k) * 32 + lane) * 2);
      ar[g] = *(const v2f*)(fragR + ((g * 8 + k) * 32 + lane) * 2);
    }
    #pragma unroll
    for (int i = 0; i < 8; ++i) {
      v2f b;
      b.x = xc[k][0][i];
      b.y = xc[k][1][i];
      nu[i] = wmma_f32_k4(an[GRADE[i]], b, nu[i]);
      xl[i] = wmma_f32_k4(al[GRADE[i]], b, xl[i]);
      xr[i] = wmma_f32_k4(ar[GRADE[i]], b, xr[i]);
    }
  }

  // biases (blade 0): nu += b_node; xl += b_left + Wl@b_node; xr += Wr@b_node
  #pragma unroll
  for (int v = 0; v < 8; ++v) {
    int ch = v + 8 * hi;
    nu[0][v] += b_node[ch];
    xl[0][v] += blp[ch];
    xr[0][v] += brp[ch];
  }

  // mv_normalize(xr, norm_a)
  #pragma unroll
  for (int v = 0; v < 8; ++v) {
    int ch = v + 8 * hi;
    float r0 = xr[0][v], r1 = xr[1][v], r2 = xr[2][v], r3 = xr[3][v];
    float r4 = xr[4][v], r5 = xr[5][v], r6 = xr[6][v], r7 = xr[7][v];
    float n0r = sqrtf(r0 * r0 + 1e-12f);
    float n1r = sqrtf(r1 * r1 + r2 * r2 + r3 * r3 + 1e-12f);
    float n2r = sqrtf(r4 * r4 + r5 * r5 + r6 * r6 + 1e-12f);
    float n3r = sqrtf(r7 * r7 + 1e-12f);
    float d0 = sigm(norm_a[ch * 4 + 0]) * (n0r - 1.f) + 1.f + 1e-6f;
    float d1 = sigm(norm_a[ch * 4 + 1]) * (n1r - 1.f) + 1.f + 1e-6f;
    float d2 = sigm(norm_a[ch * 4 + 2]) * (n2r - 1.f) + 1.f + 1e-6f;
    float d3 = sigm(norm_a[ch * 4 + 3]) * (n3r - 1.f) + 1.f + 1e-6f;
    xr[0][v] = r0 / d0;
    xr[1][v] = r1 / d1; xr[2][v] = r2 / d1; xr[3][v] = r3 / d1;
    xr[4][v] = r4 / d2; xr[5][v] = r5 / d2; xr[6][v] = r6 / d2;
    xr[7][v] = r7 / d3;
  }

  constexpr float INV_SQRT2 = 0.70710678118654752440f;
  #pragma unroll
  for (int v = 0; v < 8; ++v) {
    int ch = v + 8 * hi;
    float x[8], r[8], y[8];
    #pragma unroll
    for (int i = 0; i < 8; ++i) { x[i] = nu[i][v]; r[i] = xr[i][v]; y[i] = 0.f; }
    float w[20];
    #pragma unroll
    for (int p = 0; p < 20; ++p) w[p] = gp_w[ch * 20 + p];
    // geometric product; all table indices fold at compile time
    #pragma unroll
    for (int I = 0; I < 8; ++I)
      #pragma unroll
      for (int K = 0; K < 8; ++K)
        y[GPT.j[I * 8 + K]] += GPT.s[I * 8 + K] * w[GPT.p[I * 8 + K]] * x[I] * r[K];

    float gp[8];
    #pragma unroll
    for (int i = 0; i < 8; ++i) gp[i] = (xl[i][v] + y[i]) * INV_SQRT2;

    float i1 = gp[1] * gp[1] + gp[2] * gp[2] + gp[3] * gp[3];
    float i2 = gp[4] * gp[4] + gp[5] * gp[5] + gp[6] * gp[6];
    float i3 = gp[7] * gp[7];
    float g0 = sigm(a_n[ch * 4 + 0] * gp[0] + b_n[ch * 4 + 0]);
    float g1 = sigm(a_n[ch * 4 + 1] * i1 + b_n[ch * 4 + 1]);
    float g2 = sigm(a_n[ch * 4 + 2] * i2 + b_n[ch * 4 + 2]);
    float g3 = sigm(a_n[ch * 4 + 3] * i3 + b_n[ch * 4 + 3]);

    const float4* hb = (const float4*)(h + (size_t)node * 128 + ch * 8);
    float4 h0 = hb[0], h1 = hb[1];
    float4* op = (float4*)(out + (size_t)node * 128 + ch * 8);
    op[0] = make_float4(h0.x + g0 * gp[0], h0.y + g1 * gp[1],
                        h0.z + g1 * gp[2], h0.w + g1 * gp[3]);
    op[1] = make_float4(h1.x + g2 * gp[4], h1.y + g2 * gp[5],
                        h1.z + g2 * gp[6], h1.w + g3 * gp[7]);
  }
}

extern "C" void kernel_launch(void* const* d_in, const int* in_sizes, int n_in,
                              void* d_out, int out_size, void* d_ws, size_t ws_size,
                              hipStream_t stream) {
  const float* h       = (const float*)d_in[0];
  const int*   ei      = (const int*)  d_in[1];
  const float* W_edge  = (const float*)d_in[2];
  const float* b_edge  = (const float*)d_in[3];
  const float* a_e     = (const float*)d_in[4];
  const float* b_e     = (const float*)d_in[5];
  const float* W_node  = (const float*)d_in[6];
  const float* b_node  = (const float*)d_in[7];
  const float* W_left  = (const float*)d_in[8];
  const float* b_left  = (const float*)d_in[9];
  const float* W_right = (const float*)d_in[10];
  const float* gp_w    = (const float*)d_in[11];
  const float* norm_a  = (const float*)d_in[12];
  const float* a_n     = (const float*)d_in[13];
  const float* b_n     = (const float*)d_in[14];

  float* ws   = (float*)d_ws;
  float* agg  = ws;
  const float* fragE = ws + OFF_FRAGE;
  const float* fragN = ws + OFF_FRAGN;
  const float* fragL = ws + OFF_FRAGL;
  const float* fragR = ws + OFF_FRAGR;
  const float* blp   = ws + OFF_BLP;
  const float* brp   = ws + OFF_BRP;

  hipMemsetAsync(agg, 0, (size_t)AGG_FLOATS * sizeof(float), stream);
  egcl_pack_kernel<<<1, 32, 0, stream>>>(W_edge, W_node, W_left, W_right, b_node,
                                         b_left, ws);
  egcl_edge_kernel<<<NEDGES / 16, 32, 0, stream>>>(h, ei, fragE, b_edge, a_e, b_e, agg);
  egcl_node_kernel<<<NNODES / 16, 32, 0, stream>>>(h, agg, fragN, fragL, fragR, b_node,
                                                   blp, brp, gp_w, norm_a, a_n, b_n,
                                                   (float*)d_out);
}